// MVN_elbo_autolatent_85083302134332
// MI455X (gfx1250) — compile-verified
//
#include <hip/hip_runtime.h>
#include <hip/hip_bf16.h>
#include <math.h>

// Problem constants (fixed by the reference)
#define KAP 8
#define NN  2000
#define N0  1000
#define MM  10000
#define DD  8
#define NB  16

typedef __attribute__((ext_vector_type(2))) float v2f;
typedef __attribute__((ext_vector_type(8))) float v8f;

// ---------------------------------------------------------------------------
// Kernel 1: covariance build.  C[k][i][j] = prod_d (1+S_d) * exp(lmb[k][8]-sum S_d),
// S_d = |x1[i][d]-x2[j][d]| * exp(-lmb[k][d]).  Optional diagonal nugget.
// ---------------------------------------------------------------------------
__global__ void build_cov_kernel(const float* __restrict__ X1, int n1,
                                 const float* __restrict__ X2, int n2,
                                 const float* __restrict__ Lmb,
                                 float* __restrict__ Kout, int addNug)
{
    int k = blockIdx.y;
    long t = (long)blockIdx.x * blockDim.x + threadIdx.x;
    if (t >= (long)n1 * n2) return;
    int i = (int)(t / n2);
    int j = (int)(t % n2);
    const float* l  = Lmb + k * (DD + 1);
    const float* x1 = X1 + (size_t)i * DD;
    const float* x2 = X2 + (size_t)j * DD;
    float ssum = 0.f, prod = 1.f;
#pragma unroll
    for (int d = 0; d < DD; ++d) {
        float s = fabsf(x1[d] - x2[d]) * __expf(-l[d]);
        prod *= (1.f + s);
        ssum += s;
    }
    float v = prod * __expf(l[DD] - ssum);
    if (addNug && i == j) v += 1e-6f * __expf(l[DD]);
    Kout[(size_t)k * n1 * n2 + t] = v;
}

// ---------------------------------------------------------------------------
// Kernel 2: Cholesky panel (one workgroup per matrix k).
//   (a) factor 16x16 diagonal block in LDS (lower),
//   (b) write it back (lower part only),
//   (c) triangular-solve the panel rows below: L21 = A21 * inv(L11)^T.
// ---------------------------------------------------------------------------
__global__ void chol_panel_kernel(float* __restrict__ K, int n, int p)
{
    int k = blockIdx.x;
    float* A = K + (size_t)k * n * n;
    __shared__ float L11[NB][NB + 1];
    int tid = threadIdx.x;            // 256 threads
    int r = tid >> 4, c = tid & 15;
    L11[r][c] = A[(size_t)(p + r) * n + (p + c)];
    __syncthreads();
    for (int cc = 0; cc < NB; ++cc) {
        if (tid == 0) L11[cc][cc] = sqrtf(L11[cc][cc]);
        __syncthreads();
        if (tid > cc && tid < NB) L11[tid][cc] /= L11[cc][cc];
        __syncthreads();
        if (c > cc && r >= c) L11[r][c] -= L11[r][cc] * L11[c][cc];
        __syncthreads();
    }
    if (r >= c) A[(size_t)(p + r) * n + (p + c)] = L11[r][c];
    // panel triangular solve, one row per thread (strided)
    for (int row = p + NB + tid; row < n; row += blockDim.x) {
        float* Ar = A + (size_t)row * n + p;
        float x[NB];
#pragma unroll
        for (int cc = 0; cc < NB; ++cc) {
            float s = Ar[cc];
#pragma unroll
            for (int j = 0; j < NB; ++j)
                if (j < cc) s -= x[j] * L11[cc][j];
            x[cc] = s / L11[cc][cc];
        }
#pragma unroll
        for (int cc = 0; cc < NB; ++cc) Ar[cc] = x[cc];
    }
}

// ---------------------------------------------------------------------------
// Kernel 3: trailing SYRK update with fp32 WMMA.  One wave (32 lanes) per
// 16x16 tile of the trailing matrix: C -= L21_rows * L21_cols^T via four
// chained v_wmma_f32_16x16x4_f32 (rank-16 update). Lower tiles only.
// A-operand layout: lane(l&15)=M, K = 2*(l>>4)+{0,1} across the v2f.
// B-operand layout: lane(l&15)=N, K = 2*(l>>4)+{0,1}  -> identical addressing.
// C/D layout: vgpr v holds M = v + 8*(l>>4), N = l&15.
// ---------------------------------------------------------------------------
__global__ void chol_update_kernel(float* __restrict__ K, int n, int p)
{
    int tj = blockIdx.x, ti = blockIdx.y, k = blockIdx.z;
    if (tj > ti) return;                      // uniform per-wave exit
    float* A = K + (size_t)k * n * n;
    int t0 = p + NB;
    int ri = t0 + ti * NB;                    // tile row base
    int cj = t0 + tj * NB;                    // tile col base
    int lane = threadIdx.x;                   // 0..31 (wave32)
    int half = lane >> 4, l16 = lane & 15;
    int crow = ri + 8 * half;

    v8f cacc;
#pragma unroll
    for (int v = 0; v < 8; ++v)
        cacc[v] = A[(size_t)(crow + v) * n + (cj + l16)];

    const float* Arow = A + (size_t)(ri + l16) * n + p + 2 * half;
    const float* Brow = A + (size_t)(cj + l16) * n + p + 2 * half;
#pragma unroll
    for (int kc = 0; kc < NB; kc += 4) {
        v2f a, b;
        a.x = -Arow[kc]; a.y = -Arow[kc + 1];   // negate A: f32 WMMA has CNeg only
        b.x =  Brow[kc]; b.y =  Brow[kc + 1];
        cacc = __builtin_amdgcn_wmma_f32_16x16x4_f32(
            false, a, false, b, (short)0, cacc, false, false);
    }
#pragma unroll
    for (int v = 0; v < 8; ++v)
        A[(size_t)(crow + v) * n + (cj + l16)] = cacc[v];
}

// ---------------------------------------------------------------------------
// Kernel 4/5: blocked forward (L y = Mu) and backward (L^T x = y) solves.
// One workgroup per matrix; y kept in LDS (8 KB of 320 KB).
// ---------------------------------------------------------------------------
__global__ void fwd_solve_kernel(const float* __restrict__ K,
                                 const float* __restrict__ Mu,
                                 float* __restrict__ alpha, int n)
{
    int k = blockIdx.x;
    const float* A = K + (size_t)k * n * n;
    __shared__ float y[NN];
    __shared__ float B[NB][NB + 1];
    __shared__ float part[NB][NB + 1];
    __shared__ float off[NB];
    int tid = threadIdx.x;                 // 256
    for (int i = tid; i < n; i += 256) y[i] = Mu[(size_t)k * n + i];
    __syncthreads();
    int r = tid >> 4, c = tid & 15;
    for (int p = 0; p < n; p += NB) {
        B[r][c] = A[(size_t)(p + r) * n + (p + c)];
        float s = 0.f;
        for (int j = c; j < p; j += NB) s += A[(size_t)(p + r) * n + j] * y[j];
        part[r][c] = s;
        __syncthreads();
        if (tid < NB) {
            float o = 0.f;
#pragma unroll
            for (int t = 0; t < NB; ++t) o += part[tid][t];
            off[tid] = o;
        }
        __syncthreads();
        if (tid == 0) {
            for (int cc = 0; cc < NB; ++cc) {
                float s2 = y[p + cc] - off[cc];
                for (int j = 0; j < cc; ++j) s2 -= B[cc][j] * y[p + j];
                y[p + cc] = s2 / B[cc][cc];
            }
        }
        __syncthreads();
    }
    for (int i = tid; i < n; i += 256) alpha[(size_t)k * n + i] = y[i];
}

__global__ void bwd_solve_kernel(const float* __restrict__ K,
                                 float* __restrict__ alpha, int n)
{
    int k = blockIdx.x;
    const float* A = K + (size_t)k * n * n;
    __shared__ float x[NN];
    __shared__ float B[NB][NB + 1];
    __shared__ float part[NB][NB + 1];
    __shared__ float off[NB];
    int tid = threadIdx.x;
    for (int i = tid; i < n; i += 256) x[i] = alpha[(size_t)k * n + i];
    __syncthreads();
    int r = tid >> 4, c = tid & 15;
    for (int p = n - NB; p >= 0; p -= NB) {
        B[r][c] = A[(size_t)(p + r) * n + (p + c)];
        // off[r] = sum_{j >= p+16} L[j][p+r] * x[j]
        float s = 0.f;
        for (int j = p + NB + c; j < n; j += NB) s += A[(size_t)j * n + (p + r)] * x[j];
        part[r][c] = s;
        __syncthreads();
        if (tid < NB) {
            float o = 0.f;
#pragma unroll
            for (int t = 0; t < NB; ++t) o += part[tid][t];
            off[tid] = o;
        }
        __syncthreads();
        if (tid == 0) {
            for (int cc = NB - 1; cc >= 0; --cc) {
                float s2 = x[p + cc] - off[cc];
                for (int j = cc + 1; j < NB; ++j) s2 -= B[j][cc] * x[p + j];
                x[p + cc] = s2 / B[cc][cc];
            }
        }
        __syncthreads();
    }
    for (int i = tid; i < n; i += 256) alpha[(size_t)k * n + i] = x[i];
}

// ---------------------------------------------------------------------------
// Kernel 6: ghat[k][o] = K_nt[k][o][:] . alpha[k][:]   (one wave32 per row)
// ---------------------------------------------------------------------------
__global__ void gemv_knt_kernel(const float* __restrict__ Knt,
                                const float* __restrict__ alpha,
                                float* __restrict__ ghat)
{
    int gw = (blockIdx.x * blockDim.x + threadIdx.x) >> 5;
    int lane = threadIdx.x & 31;
    if (gw >= KAP * N0) return;
    int k = gw / N0, o = gw % N0;
    const float* row = Knt + ((size_t)k * N0 + o) * NN;
    const float* a = alpha + (size_t)k * NN;
    float s = 0.f;
    for (int j = lane; j < NN; j += 32) s += row[j] * a[j];
    for (int off = 16; off > 0; off >>= 1) s += __shfl_down(s, off, 32);
    if (lane == 0) ghat[gw] = s;
}

// ---------------------------------------------------------------------------
// Kernel 7: fhat[m][o] = psi[m] + sum_k Phi[m][k] * ghat[k][o]  (40 MB write)
// ---------------------------------------------------------------------------
__global__ void fhat_kernel(const float* __restrict__ psi,
                            const float* __restrict__ Phi,
                            const float* __restrict__ ghat,
                            float* __restrict__ out)
{
    long t = (long)blockIdx.x * blockDim.x + threadIdx.x;
    if (t >= (long)MM * N0) return;
    int m = (int)(t / N0);
    int o = (int)(t % N0);
    float s = psi[m];
#pragma unroll
    for (int k = 0; k < KAP; ++k) s += Phi[(size_t)m * KAP + k] * ghat[(size_t)k * N0 + o];
    out[t] = s;
}

// ---------------------------------------------------------------------------
extern "C" void kernel_launch(void* const* d_in, const int* in_sizes, int n_in,
                              void* d_out, int out_size, void* d_ws, size_t ws_size,
                              hipStream_t stream)
{
    const float* theta0 = (const float*)d_in[0];   // [1000,8]
    const float* Lmb    = (const float*)d_in[1];   // [8,9]
    // d_in[2] = lsigma2 (unused by the reference output)
    const float* Mu     = (const float*)d_in[3];   // [8,2000]
    const float* psi    = (const float*)d_in[4];   // [10000,1]
    const float* Phi    = (const float*)d_in[5];   // [10000,8]
    const float* theta  = (const float*)d_in[6];   // [2000,8]
    float* out = (float*)d_out;                    // [10000,1000]

    float* Ktt   = (float*)d_ws;                                   // 8*2000*2000
    float* Knt   = Ktt   + (size_t)KAP * NN * NN;                  // 8*1000*2000
    float* alpha = Knt   + (size_t)KAP * N0 * NN;                  // 8*2000
    float* ghat  = alpha + (size_t)KAP * NN;                       // 8*1000

    // 1) covariance matrices
    build_cov_kernel<<<dim3((NN * NN + 255) / 256, KAP), 256, 0, stream>>>(
        theta, NN, theta, NN, Lmb, Ktt, 1);
    build_cov_kernel<<<dim3((N0 * NN + 255) / 256, KAP), 256, 0, stream>>>(
        theta0, N0, theta, NN, Lmb, Knt, 0);

    // 2) batched blocked Cholesky (in place, lower), rank-16 WMMA updates
    for (int p = 0; p < NN; p += NB) {
        chol_panel_kernel<<<dim3(KAP), 256, 0, stream>>>(Ktt, NN, p);
        int ntr = NN - p - NB;
        if (ntr > 0) {
            int t = ntr / NB;
            chol_update_kernel<<<dim3(t, t, KAP), 32, 0, stream>>>(Ktt, NN, p);
        }
    }

    // 3) alpha = K_tt^{-1} Mu via two triangular solves
    fwd_solve_kernel<<<dim3(KAP), 256, 0, stream>>>(Ktt, Mu, alpha, NN);
    bwd_solve_kernel<<<dim3(KAP), 256, 0, stream>>>(Ktt, alpha, NN);

    // 4) ghat = K_nt @ alpha  (one wave per output row)
    gemv_knt_kernel<<<(KAP * N0 * 32 + 255) / 256, 256, 0, stream>>>(Knt, alpha, ghat);

    // 5) fhat = psi + Phi @ ghat
    fhat_kernel<<<(int)(((long)MM * N0 + 255) / 256), 256, 0, stream>>>(psi, Phi, ghat, out);
}